// BalancedBCELoss_48189533061211
// MI455X (gfx1250) — compile-verified
//
#include <hip/hip_runtime.h>
#include <hip/hip_bf16.h>

// ---------------------------------------------------------------------------
// Balanced BCE with hard-negative mining, via histogram pivot (no sort).
//   ws layout:  [0..3]   float  gsum:  0=sum(gt*m), 1=sum(1-gt), 2=pos_loss
//               [+16B]   u32    hist_cnt[NBINS]
//               [+16B+4*NBINS] float hist_sum[NBINS]
// ---------------------------------------------------------------------------

#define NBINS      4096
#define BIN_SHIFT  19          // bin = float_bits >> 19  (monotonic for v > 0)
#define P1_BLOCKS  512
#define P1_THREADS 256

typedef __attribute__((ext_vector_type(2))) float v2f;
typedef __attribute__((ext_vector_type(8))) float v8f;

__global__ void bbce_zero_ws(unsigned* __restrict__ ws, int n) {
  int i = blockIdx.x * blockDim.x + threadIdx.x;
  if (i < n) ws[i] = 0u;
}

__device__ __forceinline__ void bce_elem(float p, float g, float m,
                                         float& pos_cnt, float& neg_gt, float& pos_loss,
                                         unsigned* s_cnt, float* s_sum) {
  float lp   = fmaxf(logf(p), -100.0f);        // torch BCE log clamp
  float l1m  = fmaxf(log1pf(-p), -100.0f);
  float loss = -(g * lp + (1.0f - g) * l1m);
  float pm   = g * m;
  pos_cnt  += pm;
  neg_gt   += 1.0f - g;
  pos_loss += loss * pm;
  float nl = loss * (1.0f - g) * m;            // negative-candidate loss
  if (nl > 0.0f) {
    unsigned b = __float_as_uint(nl) >> BIN_SHIFT;
    if (b > (unsigned)(NBINS - 1)) b = NBINS - 1;
    atomicAdd(&s_cnt[b], 1u);
    atomicAdd(&s_sum[b], nl);
  }
}

__global__ void bbce_pass1(const float* __restrict__ pred,
                           const float* __restrict__ gt,
                           const float* __restrict__ masks,
                           float* __restrict__ gsum,
                           unsigned* __restrict__ gcnt,
                           float* __restrict__ gbin,
                           int N) {
  __shared__ unsigned s_cnt[NBINS];   // 16 KB
  __shared__ float    s_sum[NBINS];   // 16 KB
  __shared__ float    s_red[24];

  const int tid = threadIdx.x;
  for (int b = tid; b < NBINS; b += blockDim.x) { s_cnt[b] = 0u; s_sum[b] = 0.0f; }
  __syncthreads();

  float pos_cnt = 0.0f, neg_gt = 0.0f, pos_loss = 0.0f;

  const int gid    = blockIdx.x * blockDim.x + tid;
  const int stride = gridDim.x * blockDim.x;
  const int N4     = N >> 2;

  const float4* __restrict__ p4 = (const float4*)pred;
  const float4* __restrict__ g4 = (const float4*)gt;
  const float4* __restrict__ m4 = (const float4*)masks;

  for (int i = gid; i < N4; i += stride) {
    // gfx1250 global_prefetch_b8 for the next grid-stride tile
    __builtin_prefetch(p4 + i + stride, 0, 3);
    __builtin_prefetch(g4 + i + stride, 0, 3);
    __builtin_prefetch(m4 + i + stride, 0, 3);
    float4 p = p4[i], g = g4[i], m = m4[i];
    bce_elem(p.x, g.x, m.x, pos_cnt, neg_gt, pos_loss, s_cnt, s_sum);
    bce_elem(p.y, g.y, m.y, pos_cnt, neg_gt, pos_loss, s_cnt, s_sum);
    bce_elem(p.z, g.z, m.z, pos_cnt, neg_gt, pos_loss, s_cnt, s_sum);
    bce_elem(p.w, g.w, m.w, pos_cnt, neg_gt, pos_loss, s_cnt, s_sum);
  }
  for (int i = (N4 << 2) + gid; i < N; i += stride)
    bce_elem(pred[i], gt[i], masks[i], pos_cnt, neg_gt, pos_loss, s_cnt, s_sum);

  __syncthreads();

  // Flush privatized histogram (only touched bins -> few global atomics)
  for (int b = tid; b < NBINS; b += blockDim.x) {
    unsigned c = s_cnt[b];
    if (c) {
      atomicAdd(&gcnt[b], c);
      atomicAdd(&gbin[b], s_sum[b]);
    }
  }

  // Block reduction of the three scalar partials (wave32 shuffles + LDS)
  for (int off = 16; off > 0; off >>= 1) {
    pos_cnt  += __shfl_xor(pos_cnt,  off, 32);
    neg_gt   += __shfl_xor(neg_gt,   off, 32);
    pos_loss += __shfl_xor(pos_loss, off, 32);
  }
  const int lane = tid & 31, wave = tid >> 5;
  if (lane == 0) { s_red[wave] = pos_cnt; s_red[8 + wave] = neg_gt; s_red[16 + wave] = pos_loss; }
  __syncthreads();
  if (tid == 0) {
    float a = 0.0f, b = 0.0f, c = 0.0f;
    const int nw = (blockDim.x + 31) >> 5;
    for (int w = 0; w < nw; ++w) { a += s_red[w]; b += s_red[8 + w]; c += s_red[16 + w]; }
    atomicAdd(&gsum[0], a);
    atomicAdd(&gsum[1], b);
    atomicAdd(&gsum[2], c);
  }
}

__global__ void bbce_pass2(const float* __restrict__ gsum,
                           const unsigned* __restrict__ gcnt,
                           const float* __restrict__ gbin,
                           float* __restrict__ out) {
  __shared__ float    s_sum[NBINS];   // 16 KB
  __shared__ unsigned s_cnt[NBINS];   // 16 KB
  __shared__ float    s_sc[2];

  const int tid = threadIdx.x;
  for (int b = tid; b < NBINS; b += blockDim.x) { s_sum[b] = gbin[b]; s_cnt[b] = gcnt[b]; }
  __syncthreads();

  if (tid == 0) {
    float num_pos = floorf(gsum[0]);
    float num_neg = floorf(fminf(gsum[1], num_pos * 3.0f));
    // Descending scan: keep whole bins while count fits; fractional boundary bin.
    float remaining = num_neg;
    for (int b = NBINS - 1; b >= 0; --b) {
      float c = (float)s_cnt[b];
      if (c <= remaining) {
        remaining -= c;                 // take whole bin
      } else {
        s_sum[b] *= remaining / c;      // fractional boundary bin (c >= 1 here)
        for (int j = b - 1; j >= 0; --j) s_sum[j] = 0.0f;
        break;
      }
    }
    s_sc[0] = num_pos;
    s_sc[1] = num_neg;
  }
  __syncthreads();

  // Wave 0 (EXEC all-ones on wave32) reduces the 4096 pivoted bin-sums.
  if (tid < 32) {
    float total;
#if __has_builtin(__builtin_amdgcn_wmma_f32_16x16x4_f32)
    // D = A(16x4) * ones(4x16) + C : every D[m][n] accumulates rowsum(A,m).
    // Sum of all 256 D entries == 16 * sum(A)  ->  total = (sum D)/16.
    v8f acc = {};
    v2f bones; bones.x = 1.0f; bones.y = 1.0f;
    for (int base = 0; base < NBINS; base += 64) {
      v2f a;
      a.x = s_sum[base + tid * 2];
      a.y = s_sum[base + tid * 2 + 1];
      acc = __builtin_amdgcn_wmma_f32_16x16x4_f32(
          /*neg_a=*/false, a, /*neg_b=*/false, bones,
          /*c_mod=*/(short)0, acc, /*reuse_a=*/false, /*reuse_b=*/false);
    }
    float s = 0.0f;
#pragma unroll
    for (int i = 0; i < 8; ++i) s += acc[i];
    for (int off = 16; off > 0; off >>= 1) s += __shfl_xor(s, off, 32);
    total = s * 0.0625f;  // / 16
#else
    float s = 0.0f;
    for (int b = tid; b < NBINS; b += 32) s += s_sum[b];
    for (int off = 16; off > 0; off >>= 1) s += __shfl_xor(s, off, 32);
    total = s;
#endif
    if (tid == 0) {
      float positive_loss = gsum[2];
      out[0] = (positive_loss + total) / (s_sc[0] + s_sc[1] + 1e-6f);
    }
  }
}

extern "C" void kernel_launch(void* const* d_in, const int* in_sizes, int n_in,
                              void* d_out, int out_size, void* d_ws, size_t ws_size,
                              hipStream_t stream) {
  (void)n_in; (void)out_size; (void)ws_size;
  const float* pred  = (const float*)d_in[0];
  const float* gt    = (const float*)d_in[1];
  const float* masks = (const float*)d_in[2];
  float* out = (float*)d_out;
  const int N = in_sizes[0];

  float*    gsum = (float*)d_ws;
  unsigned* gcnt = (unsigned*)((char*)d_ws + 16);
  float*    gbin = (float*)((char*)d_ws + 16 + 4 * NBINS);

  const int nzero = 4 + 2 * NBINS;
  bbce_zero_ws<<<(nzero + 255) / 256, 256, 0, stream>>>((unsigned*)d_ws, nzero);
  bbce_pass1<<<P1_BLOCKS, P1_THREADS, 0, stream>>>(pred, gt, masks, gsum, gcnt, gbin, N);
  bbce_pass2<<<1, 256, 0, stream>>>(gsum, gcnt, gbin, out);
}